// WindowNetwork_80977313399300
// MI455X (gfx1250) — compile-verified
//
#include <hip/hip_runtime.h>
#include <math.h>

#define NG 20
#define NL 80
#define HID 400
#define BATCH 1024
#define UMAX 1024
#define OUTC 60
#define OUTC_PAD 64
#define ATT_SCALE 0.05f

typedef __attribute__((ext_vector_type(2))) float v2f;
typedef __attribute__((ext_vector_type(8))) float v8f;

// ---------------------------------------------------------------------------
// Kernel 1: out[1024,60] = y0 @ W^T + bias, fused with exp / exp / exp*s+prev_k
// One wave computes one 16x16 tile via V_WMMA_F32_16X16X4_F32 (K-loop = 100).
// Padded columns 60..63 read a clamped (valid) W row unconditionally; their
// accumulator lanes are dead and never stored, so there is NO divergence in
// the hot loop: body = 2x global_load_b64 + v_wmma per K-step.
// Block = 128 threads = 4 waves -> covers the 64 (padded) output columns.
// Grid  = 64 blocks   -> covers the 1024 batch rows.
// ---------------------------------------------------------------------------
__global__ __launch_bounds__(128)
void window_gemm_act_kernel(const float* __restrict__ Y,     // [B, H]
                            const float* __restrict__ Wm,    // [60, H] row-major
                            const float* __restrict__ bias,  // [60]
                            const float* __restrict__ prevk, // [B, 20]
                            float* __restrict__ params,      // ws: [B, 64] (a|b|k|pad)
                            float* __restrict__ out_k)       // [B, 20]
{
    const int lane = threadIdx.x & 31;
    const int wave = threadIdx.x >> 5;          // column tile 0..3
    const int half = lane >> 4;                 // 0 / 1 (K sub-pair select)
    const int r    = lane & 15;

    const int arow = blockIdx.x * 16 + r;       // A-matrix row (batch)
    const int col  = wave * 16 + r;             // B-matrix column (output ch)
    const int colc = (col < OUTC) ? col : (OUTC - 1);   // clamped, branch-free

    // Loop-invariant bases (include the half-dependent +2h element offset so
    // the unrolled loop uses pure immediate offsets).
    const float* __restrict__ aptr = Y  + arow * HID + 2 * half;
    const float* __restrict__ bptr = Wm + colc * HID + 2 * half;

    v8f acc = {};
    // A 16x4 f32: lane(half,r): {A[r][k+2h], A[r][k+2h+1]}  (VGPR0/1 pair)
    // B 4x16 f32: lane(half,r): {W[col][k+2h], W[col][k+2h+1]} (mirrored)
    #pragma unroll 8
    for (int kk = 0; kk < HID; kk += 4) {
        v2f av = *(const v2f*)(aptr + kk);
        v2f bv = *(const v2f*)(bptr + kk);
        acc = __builtin_amdgcn_wmma_f32_16x16x4_f32(
            /*neg_a=*/false, av, /*neg_b=*/false, bv,
            /*c_mod=*/(short)0, acc, /*reuse_a=*/false, /*reuse_b=*/false);
    }

    // C/D tile layout: VGPR v, lanes 0-15 -> M=v, N=lane; lanes 16-31 -> M=v+8.
    const float bs = bias[colc];                // dead value for padded cols
    #pragma unroll
    for (int v = 0; v < 8; ++v) {
        const int m   = blockIdx.x * 16 + v + 8 * half;  // batch row
        const float x = acc[v] + bs;
        float res;
        if (col < 2 * NG) {                 // a (0..19) and b (20..39): exp
            res = __expf(x);
        } else if (col < OUTC) {            // k (40..59): exp*scale + prev_k
            res = __expf(x) * ATT_SCALE + prevk[m * NG + (col - 2 * NG)];
            out_k[m * NG + (col - 2 * NG)] = res;
        } else {
            res = 0.0f;                     // padded columns
        }
        params[m * OUTC_PAD + col] = res;
    }
}

// ---------------------------------------------------------------------------
// Kernel 2: per batch row b: phi[b,u] = mask * sum_n a_n * exp(-b_n*(k_n-u)^2)
// and w_t[b, letter] scatter-add via an LDS histogram (ds_add_f32).
// One 256-thread block per row; a/b/k broadcast through LDS.
// ---------------------------------------------------------------------------
__global__ __launch_bounds__(256)
void window_phi_scatter_kernel(const int* __restrict__ text,    // [B, U]
                               const int* __restrict__ tlen,    // [B]
                               const float* __restrict__ params,// [B, 64]
                               float* __restrict__ out_wt,      // [B, 80]
                               float* __restrict__ out_phi)     // [B, U]
{
    __shared__ float sa[NG];
    __shared__ float sb[NG];
    __shared__ float sk[NG];
    __shared__ float swt[NL];

    const int b   = blockIdx.x;
    const int tid = threadIdx.x;

    if (tid < OUTC) {
        const float v = params[b * OUTC_PAD + tid];
        if (tid < NG)            sa[tid]          = v;
        else if (tid < 2 * NG)   sb[tid - NG]     = v;
        else                     sk[tid - 2 * NG] = v;
    }
    if (tid < NL) swt[tid] = 0.0f;
    __syncthreads();

    const int len = tlen[b];
    const int* __restrict__ trow = text + b * UMAX;
    float* __restrict__ prow = out_phi + b * UMAX;

    for (int u = tid; u < UMAX; u += 256) {
        float phi = 0.0f;
        if (u < len) {
            const float uf = (float)u;
            #pragma unroll
            for (int n = 0; n < NG; ++n) {
                const float t = sk[n] - uf;
                phi += sa[n] * __expf(-sb[n] * t * t);
            }
        }
        prow[u] = phi;                       // masked positions write 0
        if (u < len) {
            atomicAdd(&swt[trow[u]], phi);   // ds_add_f32 histogram
        }
    }
    __syncthreads();
    if (tid < NL) out_wt[b * NL + tid] = swt[tid];
}

// ---------------------------------------------------------------------------
// d_in order: text_inputs, text_lengths, y_0, prev_k, W, bias
// d_out: w_t [B*80] ++ k [B*20] ++ phi [B*U]  (float32)
// ---------------------------------------------------------------------------
extern "C" void kernel_launch(void* const* d_in, const int* in_sizes, int n_in,
                              void* d_out, int out_size, void* d_ws, size_t ws_size,
                              hipStream_t stream) {
    const int*   text  = (const int*)d_in[0];
    const int*   tlen  = (const int*)d_in[1];
    const float* y0    = (const float*)d_in[2];
    const float* prevk = (const float*)d_in[3];
    const float* Wm    = (const float*)d_in[4];
    const float* bias  = (const float*)d_in[5];

    float* out     = (float*)d_out;
    float* out_wt  = out;                                  // [B, 80]
    float* out_k   = out + (size_t)BATCH * NL;             // [B, 20]
    float* out_phi = out + (size_t)BATCH * (NL + NG);      // [B, U]
    float* params  = (float*)d_ws;                         // [B, 64] scratch

    window_gemm_act_kernel<<<BATCH / 16, 128, 0, stream>>>(
        y0, Wm, bias, prevk, params, out_k);

    window_phi_scatter_kernel<<<BATCH, 256, 0, stream>>>(
        text, tlen, params, out_wt, out_phi);
}